// SeparableAttention_56315611185762
// MI455X (gfx1250) — compile-verified
//
#include <hip/hip_runtime.h>
#include <hip/hip_bf16.h>

typedef __attribute__((ext_vector_type(2))) float v2f;
typedef __attribute__((ext_vector_type(8))) float v8f;
typedef __attribute__((ext_vector_type(4))) unsigned int v4u;
typedef __attribute__((ext_vector_type(4))) int v4i;
typedef __attribute__((ext_vector_type(8))) int v8i;

constexpr int BATCH = 2;
constexpr int CH    = 512;
constexpr int SEQ   = 2048;          // H*W = 64*32
constexpr int QKC   = 576;           // C + freq(32) + time(32)
constexpr int INNER = 512;
constexpr int NHEAD = 8;
constexpr int HD    = 64;
constexpr float TWO_PI_F = 6.283185307179586f;

__device__ __forceinline__ v8f vzero8() {
  v8f z;
#pragma unroll
  for (int i = 0; i < 8; ++i) z[i] = 0.0f;
  return z;
}

// D = A(16x4, f32) * B(4x16, f32) + C(16x16, f32)  -- exact fp32 matrix-pipe op
__device__ __forceinline__ v8f wmma4(v2f a, v2f b, v8f c) {
  return __builtin_amdgcn_wmma_f32_16x16x4_f32(false, a, false, b, (short)0, c,
                                               false, false);
}

// ---------------------------------------------------------------------------
// Tensor Data Mover: 2D f32 tile load (global -> LDS) with LDS padding to
// create a bank-conflict-free pitch.  D# built per CDNA5 ISA
// (CDNA5 ISA): group0 = {flags, lds_addr, gaddr_lo,
// gaddr_hi|type=2}; group1 packs data_size/pad + dims/strides.
// This toolchain's builtin takes 6 args (g0, g1, g2, g3, g4, cpol).
// Tracked with TENSORcnt; issue from a single wave.
// ---------------------------------------------------------------------------
__device__ __forceinline__ void tdm_load_2d_f32(
    unsigned int lds_addr, const float* gptr, unsigned int tile_d0,
    unsigned int tile_d1, unsigned long long stride0_elems,
    unsigned int pad_interval_code, unsigned int pad_amount_code) {
  const unsigned long long ga = (unsigned long long)(uintptr_t)gptr;
  v4u g0;
  g0[0] = 1u;  // count=1 (valid user descriptor), no gather, no restore
  g0[1] = lds_addr;
  g0[2] = (unsigned int)(ga & 0xffffffffu);
  g0[3] = (unsigned int)((ga >> 32) & 0x01ffffffu) | (2u << 30);  // type=2
  const unsigned int tensor_d0 = tile_d0;  // exact-fit tensor: no OOB
  const unsigned int tensor_d1 = tile_d1;
  v8i g1;
  g1[0] = (int)((2u << 16)                     // data_size = 4 bytes
                | (1u << 20)                   // pad_enable
                | (pad_interval_code << 22)    // dwords between pads
                | (pad_amount_code << 25));    // pad dwords
  g1[1] = (int)((tensor_d0 & 0xffffu) << 16);  // atomic_barrier_addr = 0
  g1[2] = (int)((tensor_d0 >> 16) | ((tensor_d1 & 0xffffu) << 16));
  g1[3] = (int)((tensor_d1 >> 16) | (tile_d0 << 16));
  g1[4] = (int)(tile_d1 & 0xffffu);            // tile_dim2 = 0 (2D)
  g1[5] = (int)(unsigned int)(stride0_elems & 0xffffffffu);
  g1[6] = (int)(unsigned int)((stride0_elems >> 32) & 0xffffu);  // dim1_stride=0
  g1[7] = 0;
  const v4i z4 = {0, 0, 0, 0};  // groups 2/3 unused for 2D tensors
  v8i z8;
#pragma unroll
  for (int i = 0; i < 8; ++i) z8[i] = 0;
  __builtin_amdgcn_tensor_load_to_lds(g0, g1, z4, z4, z8, 0);
}

// ---------------------------------------------------------------------------
// Kernel 1: Q/K/V projections.  GEMM M=SEQ, N=INNER, K=576 (Q,K) / 512 (V).
// Positional embedding channels (k in [512,576)) are synthesized on the fly.
// LDS layouts chosen so every WMMA operand pair is one aligned ds_load_b64:
//   Alds[m][k] pitch 18, Blds[n][k] pitch 18.
// grid = (16*8, 3, BATCH), block = 256 (8 waves, 4(M) x 2(N), 32x32 per wave)
// ---------------------------------------------------------------------------
__global__ __launch_bounds__(256) void qkv_proj_kernel(
    const float* __restrict__ hidden, const float* __restrict__ Wq,
    const float* __restrict__ Wk, const float* __restrict__ Wv,
    float* __restrict__ Qb, float* __restrict__ Kb, float* __restrict__ Vb) {
  constexpr int BM = 128, BN = 64, KC = 16, KP = 18;
  const int nt = blockIdx.x & 7;   // INNER/BN = 8
  const int mt = blockIdx.x >> 3;  // SEQ/BM  = 16
  const int which = blockIdx.y;    // 0=Q 1=K 2=V
  const int b = blockIdx.z;

  const float* Wmat = (which == 0) ? Wq : (which == 1) ? Wk : Wv;
  float* Out        = (which == 0) ? Qb : (which == 1) ? Kb : Vb;
  const int Kdim    = (which == 2) ? CH : QKC;

  __shared__ float Alds[BM][KP];
  __shared__ float Blds[BN][KP];

  const int tid = threadIdx.x;
  const int lane = tid & 31;
  const int lo = lane & 15, hi = lane >> 4;
  const int wm = (tid >> 5) >> 1;  // 0..3
  const int wn = (tid >> 5) & 1;   // 0..1
  const int m0 = mt * BM, n0 = nt * BN;
  const float* hb = hidden + (size_t)b * CH * SEQ;

  v8f acc[2][2];
#pragma unroll
  for (int im = 0; im < 2; ++im)
#pragma unroll
    for (int in = 0; in < 2; ++in) acc[im][in] = vzero8();

  for (int k0 = 0; k0 < Kdim; k0 += KC) {
    // A tile: Alds[mm][kk] = X'[m0+mm][k0+kk]; hidden is [c][s]: coalesced
    // over s.  Embedding channels computed inline.
    for (int i = tid; i < KC * BM; i += 256) {
      const int mm = i & (BM - 1);
      const int kk = i >> 7;
      const int k = k0 + kk;
      const int s = m0 + mm;
      float v;
      if (k < CH) {
        v = hb[(size_t)k * SEQ + s];
      } else {
        const int e = k - CH;
        if (e < 32) {  // freq embedding over height axis
          const int f = e >> 1;
          const int hrow = s >> 5;  // h = s / W
          const int idx = hrow * 16 + f;
          const float lx = (idx == 0) ? -2.0f : __log2f((float)idx);
          const float th = lx * (float)(f + 1);
          v = (e & 1) ? __sinf(th) : __cosf(th);
        } else {  // time embedding over width axis
          const int e2 = e - 32;
          const int t = e2 >> 1;
          const int w = s & 31;  // w = s % W
          const float th =
              ((float)t + 0.5f) * (float)(w * 16 + t) * (TWO_PI_F / 512.0f);
          v = (e2 & 1) ? __sinf(th) : __cosf(th);
        }
      }
      Alds[mm][kk] = v;
    }
    // B tile transposed: Blds[nn][kk] = W[k0+kk][n0+nn]
    for (int i = tid; i < KC * BN; i += 256) {
      const int nn = i & (BN - 1);
      const int kk = i >> 6;
      Blds[nn][kk] = Wmat[(size_t)(k0 + kk) * INNER + n0 + nn];
    }
    if (k0 + KC < Kdim && k0 + KC < CH)  // prefetch next tiles into L2/L0
      __builtin_prefetch(&hb[(size_t)(k0 + KC) * SEQ + m0 + (tid & 127)], 0, 1);
    if (k0 + KC < Kdim)
      __builtin_prefetch(&Wmat[(size_t)(k0 + KC) * INNER + n0 + (tid & 63)], 0, 1);
    __syncthreads();
#pragma unroll
    for (int kk = 0; kk < KC; kk += 4) {
      v2f af[2], bf[2];
#pragma unroll
      for (int im = 0; im < 2; ++im) {
        const int row = wm * 32 + im * 16 + lo;
        af[im] = *reinterpret_cast<const v2f*>(&Alds[row][kk + 2 * hi]);
      }
#pragma unroll
      for (int in = 0; in < 2; ++in) {
        const int col = wn * 32 + in * 16 + lo;
        bf[in] = *reinterpret_cast<const v2f*>(&Blds[col][kk + 2 * hi]);
      }
#pragma unroll
      for (int im = 0; im < 2; ++im)
#pragma unroll
        for (int in = 0; in < 2; ++in)
          acc[im][in] = wmma4(af[im], bf[in], acc[im][in]);
    }
    __syncthreads();
  }

  // store head-major [b,h,s,d]; component r <-> row (.. + r + 8*hi), col lo
#pragma unroll
  for (int im = 0; im < 2; ++im)
#pragma unroll
    for (int in = 0; in < 2; ++in) {
      const int n = n0 + wn * 32 + in * 16 + lo;
      const int h = n >> 6, d = n & 63;
#pragma unroll
      for (int r = 0; r < 8; ++r) {
        const int srow = m0 + wm * 32 + im * 16 + r + 8 * hi;
        Out[(((size_t)b * NHEAD + h) * SEQ + srow) * HD + d] = acc[im][in][r];
      }
    }
}

// ---------------------------------------------------------------------------
// Kernel 2: flash attention.  One WG per (b, h, 64 q-rows); 4 waves, each
// owns a 16-row q strip.  K tiles are fetched by the Tensor Data Mover into
// double-buffered LDS (TDM inserts 2-dword padding every 64 dwords -> pitch
// 66, bank-conflict-free, b64-aligned); tile j+1 is in flight while tile j
// is consumed (s_wait_tensorcnt 1).  V is staged transposed; P is reshaped
// D-layout -> A-fragment layout via a private per-wave LDS strip (LDS is
// DScnt-in-order within a wave).
// grid = (SEQ/64, NHEAD, BATCH), block = 128
// ---------------------------------------------------------------------------
__global__ __launch_bounds__(128) void flash_attn_kernel(
    const float* __restrict__ Qb, const float* __restrict__ Kb,
    const float* __restrict__ Vb, float* __restrict__ Ob) {
  __shared__ float Kr[2][64][66];   // row-major K tiles (TDM destination)
  __shared__ float VtT[HD][66];     // transposed V tile: VtT[d][kj]
  __shared__ float Pl[4][16][66];   // per-wave P strip

  const int qt = blockIdx.x, h = blockIdx.y, b = blockIdx.z;
  const int bh = b * NHEAD + h;
  const float* Q = Qb + (size_t)bh * SEQ * HD;
  const float* K = Kb + (size_t)bh * SEQ * HD;
  const float* V = Vb + (size_t)bh * SEQ * HD;

  const int tid = threadIdx.x;
  const int wave = tid >> 5;
  const int lane = tid & 31;
  const int lo = lane & 15, hi = lane >> 4;
  const int q0 = qt * 64 + wave * 16;

  // Q strip (16 rows x 64) as 16 A-fragments, kept in registers (b64 loads)
  v2f qf[16];
#pragma unroll
  for (int c = 0; c < 16; ++c)
    qf[c] = *reinterpret_cast<const v2f*>(Q + (size_t)(q0 + lo) * HD + 4 * c +
                                          2 * hi);

  v8f oacc[4];
#pragma unroll
  for (int n = 0; n < 4; ++n) oacc[n] = vzero8();
  float rmax[8], rsum[8];
#pragma unroll
  for (int r = 0; r < 8; ++r) {
    rmax[r] = -1e30f;
    rsum[r] = 0.0f;
  }

  const float scale = 0.125f;  // 1/sqrt(64)

  // prologue: TDM fetch of K tile 0 (pad: every 64 dwords add 2 -> pitch 66)
  if (wave == 0)
    tdm_load_2d_f32((unsigned int)(uintptr_t)&Kr[0][0][0], K, 64, 64, 64,
                    /*interval 64 dw*/ 5, /*amount 2 dw*/ 1);

  for (int it = 0; it < SEQ / 64; ++it) {
    const int j0 = it * 64;
    // V tile, transposed store (float4 global reads)
    for (int i = tid; i < 64 * 16; i += 128) {
      const int kj = i >> 4;
      const int d4 = (i & 15) * 4;
      const float4 vv = *reinterpret_cast<const float4*>(
          V + (size_t)(j0 + kj) * HD + d4);
      VtT[d4 + 0][kj] = vv.x;
      VtT[d4 + 1][kj] = vv.y;
      VtT[d4 + 2][kj] = vv.z;
      VtT[d4 + 3][kj] = vv.w;
    }
    if (wave == 0) {
      if (j0 + 64 < SEQ) {  // keep next K tile in flight while computing
        tdm_load_2d_f32((unsigned int)(uintptr_t)&Kr[(it + 1) & 1][0][0],
                        K + (size_t)(j0 + 64) * HD, 64, 64, 64, 5, 1);
        __builtin_amdgcn_s_wait_tensorcnt(1);
      } else {
        __builtin_amdgcn_s_wait_tensorcnt(0);
      }
    }
    __syncthreads();
    const float(*Kc)[66] = Kr[it & 1];

    // scores = Q * K^T  (16x64 strip per wave, 64 wmma)
    v8f sc[4];
#pragma unroll
    for (int n = 0; n < 4; ++n) sc[n] = vzero8();
#pragma unroll
    for (int c = 0; c < 16; ++c) {
#pragma unroll
      for (int n = 0; n < 4; ++n) {
        const v2f bf =
            *reinterpret_cast<const v2f*>(&Kc[n * 16 + lo][4 * c + 2 * hi]);
        sc[n] = wmma4(qf[c], bf, sc[n]);
      }
    }

    // online softmax: component r holds row (r + 8*hi), lane lo = column
#pragma unroll
    for (int r = 0; r < 8; ++r) {
      float lm =
          fmaxf(fmaxf(sc[0][r], sc[1][r]), fmaxf(sc[2][r], sc[3][r])) * scale;
#pragma unroll
      for (int mask = 1; mask < 16; mask <<= 1)
        lm = fmaxf(lm, __shfl_xor(lm, mask, 32));
      const float nm = fmaxf(rmax[r], lm);
      const float corr = __expf(rmax[r] - nm);
      rmax[r] = nm;
      rsum[r] *= corr;
#pragma unroll
      for (int n = 0; n < 4; ++n) oacc[n][r] *= corr;
      float ps = 0.0f;
#pragma unroll
      for (int n = 0; n < 4; ++n) {
        const float p = __expf(sc[n][r] * scale - nm);
        sc[n][r] = p;
        ps += p;
      }
#pragma unroll
      for (int mask = 1; mask < 16; mask <<= 1) ps += __shfl_xor(ps, mask, 32);
      rsum[r] += ps;
    }

    // reshape P (D-layout) into A-fragment layout through private LDS strip
#pragma unroll
    for (int n = 0; n < 4; ++n)
#pragma unroll
      for (int r = 0; r < 8; ++r) Pl[wave][r + 8 * hi][n * 16 + lo] = sc[n][r];

    // O += P * V  (64 wmma)
#pragma unroll
    for (int c = 0; c < 16; ++c) {
      const v2f pf =
          *reinterpret_cast<const v2f*>(&Pl[wave][lo][4 * c + 2 * hi]);
#pragma unroll
      for (int n = 0; n < 4; ++n) {
        const v2f vf =
            *reinterpret_cast<const v2f*>(&VtT[n * 16 + lo][4 * c + 2 * hi]);
        oacc[n] = wmma4(pf, vf, oacc[n]);
      }
    }
    __syncthreads();
  }

  // normalize and store O as [b, s, h*64+d] for the output projection
#pragma unroll
  for (int n = 0; n < 4; ++n) {
    const int d = n * 16 + lo;
#pragma unroll
    for (int r = 0; r < 8; ++r) {
      const int s = q0 + r + 8 * hi;
      Ob[((size_t)b * SEQ + s) * INNER + h * HD + d] = oacc[n][r] / rsum[r];
    }
  }
}

// ---------------------------------------------------------------------------
// Kernel 3: output projection + bias + residual.  GEMM M=SEQ, N=CH, K=512.
// Alds[m][k] / Blds[n][k] pitch-18 layouts (aligned b64 fragment loads);
// C tile staged through LDS so the [b,c,s] store is coalesced along s.
// grid = (16*8, 1, BATCH), block = 256
// ---------------------------------------------------------------------------
__global__ __launch_bounds__(256) void out_proj_kernel(
    const float* __restrict__ Ob, const float* __restrict__ Wout,
    const float* __restrict__ bout, const float* __restrict__ hidden,
    float* __restrict__ out) {
  constexpr int BM = 128, BN = 64, KC = 16, KP = 18;
  const int nt = blockIdx.x & 7;
  const int mt = blockIdx.x >> 3;
  const int b = blockIdx.z;

  __shared__ float Alds[BM][KP];
  __shared__ float Blds[BN][KP];
  __shared__ float Clds[BN][BM + 4];

  const int tid = threadIdx.x;
  const int lane = tid & 31;
  const int lo = lane & 15, hi = lane >> 4;
  const int wm = (tid >> 5) >> 1;
  const int wn = (tid >> 5) & 1;
  const int m0 = mt * BM, n0 = nt * BN;

  v8f acc[2][2];
#pragma unroll
  for (int im = 0; im < 2; ++im)
#pragma unroll
    for (int in = 0; in < 2; ++in) acc[im][in] = vzero8();

  for (int k0 = 0; k0 < INNER; k0 += KC) {
    // A = Ob[s][k] row-major: natural copy into Alds[m][k]
    for (int i = tid; i < KC * BM; i += 256) {
      const int kk = i & (KC - 1);
      const int mm = i >> 4;
      Alds[mm][kk] = Ob[((size_t)b * SEQ + m0 + mm) * INNER + k0 + kk];
    }
    for (int i = tid; i < KC * BN; i += 256) {
      const int nn = i & (BN - 1);
      const int kk = i >> 6;
      Blds[nn][kk] = Wout[(size_t)(k0 + kk) * CH + n0 + nn];
    }
    if (k0 + KC < INNER)
      __builtin_prefetch(&Wout[(size_t)(k0 + KC) * CH + n0 + (tid & 63)], 0, 1);
    __syncthreads();
#pragma unroll
    for (int kk = 0; kk < KC; kk += 4) {
      v2f af[2], bf[2];
#pragma unroll
      for (int im = 0; im < 2; ++im) {
        const int row = wm * 32 + im * 16 + lo;
        af[im] = *reinterpret_cast<const v2f*>(&Alds[row][kk + 2 * hi]);
      }
#pragma unroll
      for (int in = 0; in < 2; ++in) {
        const int col = wn * 32 + in * 16 + lo;
        bf[in] = *reinterpret_cast<const v2f*>(&Blds[col][kk + 2 * hi]);
      }
#pragma unroll
      for (int im = 0; im < 2; ++im)
#pragma unroll
        for (int in = 0; in < 2; ++in)
          acc[im][in] = wmma4(af[im], bf[in], acc[im][in]);
    }
    __syncthreads();
  }

  // stage C tile -> LDS, then write [b,c,s] coalesced with bias + residual
#pragma unroll
  for (int im = 0; im < 2; ++im)
#pragma unroll
    for (int in = 0; in < 2; ++in) {
      const int nn = wn * 32 + in * 16 + lo;
#pragma unroll
      for (int r = 0; r < 8; ++r) {
        const int mm = wm * 32 + im * 16 + r + 8 * hi;
        Clds[nn][mm] = acc[im][in][r];
      }
    }
  __syncthreads();
  for (int i = tid; i < BN * BM; i += 256) {
    const int ss = i & (BM - 1);
    const int nn = i >> 7;
    const int c = n0 + nn;
    const size_t gi = ((size_t)b * CH + c) * SEQ + m0 + ss;
    out[gi] = Clds[nn][ss] + bout[c] + hidden[gi];
  }
}

// ---------------------------------------------------------------------------
extern "C" void kernel_launch(void* const* d_in, const int* in_sizes, int n_in,
                              void* d_out, int out_size, void* d_ws,
                              size_t ws_size, hipStream_t stream) {
  (void)in_sizes; (void)n_in; (void)out_size; (void)ws_size;
  const float* hidden = (const float*)d_in[0];
  const float* Wq     = (const float*)d_in[1];
  const float* Wk     = (const float*)d_in[2];
  const float* Wv     = (const float*)d_in[3];
  const float* Wout   = (const float*)d_in[4];
  const float* bout   = (const float*)d_in[5];
  float* out = (float*)d_out;

  // workspace: Q, K, V (head-major) + O (token-major) = 4 * 8 MB = 32 MB
  float* ws = (float*)d_ws;
  const size_t chunk = (size_t)BATCH * SEQ * INNER;  // 2,097,152 floats
  float* Qb = ws;
  float* Kb = ws + chunk;
  float* Vb = ws + 2 * chunk;
  float* Ob = ws + 3 * chunk;

  qkv_proj_kernel<<<dim3(128, 3, BATCH), 256, 0, stream>>>(hidden, Wq, Wk, Wv,
                                                           Qb, Kb, Vb);
  flash_attn_kernel<<<dim3(SEQ / 64, NHEAD, BATCH), 128, 0, stream>>>(Qb, Kb,
                                                                      Vb, Ob);
  out_proj_kernel<<<dim3(128, 1, BATCH), 256, 0, stream>>>(Ob, Wout, bout,
                                                           hidden, out);
}